// Proposed_model_6820408066255
// MI455X (gfx1250) — compile-verified
//
#include <hip/hip_runtime.h>
#include <hip/hip_bf16.h>

typedef __attribute__((ext_vector_type(2))) float v2f;
typedef __attribute__((ext_vector_type(8))) float v8f;
typedef int v4i_vs __attribute__((vector_size(16)));  // matches async-LDS builtin param

#define HID 64

#if defined(__gfx1250__) && __has_builtin(__builtin_amdgcn_global_load_async_to_lds_b128)
#define USE_ASYNC_LDS 1
#else
#define USE_ASYNC_LDS 0
#endif

// ---------------------------------------------------------------------------
// Utility kernels
// ---------------------------------------------------------------------------
__global__ void fill_f32(float* __restrict__ p, long n, float v) {
    for (long t = blockIdx.x * (long)blockDim.x + threadIdx.x; t < n;
         t += (long)gridDim.x * blockDim.x)
        p[t] = v;
}

__global__ void deg_count(const int* __restrict__ ids, int E, float* __restrict__ deg) {
    int e = blockIdx.x * blockDim.x + threadIdx.x;
    if (e < E) atomicAdd(&deg[ids[e]], 1.0f);
}

// mode 1: p = 1/sqrt(max(p,1));  mode 0: p = 1/max(p,1)
__global__ void recip_kernel(float* __restrict__ p, int n, int use_sqrt) {
    int i = blockIdx.x * blockDim.x + threadIdx.x;
    if (i >= n) return;
    float d = fmaxf(p[i], 1.0f);
    p[i] = use_sqrt ? (1.0f / sqrtf(d)) : (1.0f / d);
}

// out[dst[e],f] += h[src[e],f] * (sscale?sscale[src[e]]:1) * (escale?escale[e]:1)
// 64 consecutive threads cooperate on one edge (coalesced rows).
__global__ void scatter_add(const float* __restrict__ h, const int* __restrict__ src,
                            const int* __restrict__ dst, const float* __restrict__ sscale,
                            const float* __restrict__ escale, float* __restrict__ out,
                            long n_ef) {
    for (long t = blockIdx.x * (long)blockDim.x + threadIdx.x; t < n_ef;
         t += (long)gridDim.x * blockDim.x) {
        long e = t >> 6;
        int f = (int)(t & 63);
        int s = src[e];
        int d = dst[e];
        float v = h[(size_t)s * HID + f];
        if (sscale) v *= sscale[s];
        if (escale) v *= escale[e];
        atomicAdd(&out[(size_t)d * HID + f], v);
    }
}

// ---------------------------------------------------------------------------
// WMMA GEMM: Y[m,n] = base + alpha * rs[m] * sum_k X[m,k] * W[k,n]
//   base = accumulate ? Y[m,n] : (bias ? bias[n] : 0)
// X:[n_rows,64], W:[64,64] row-major, staged in LDS once per block.
// One wave computes 16 rows x 64 cols: one A fragment feeds 4 WMMA chains
// (4x A-reuse). 256 threads = 8 waves = 128 rows per block.
// ---------------------------------------------------------------------------
__global__ void __launch_bounds__(256)
gemm64_wmma(const float* __restrict__ X, const float* __restrict__ rowscale,
            const float* __restrict__ Wg, const float* __restrict__ bias,
            float* __restrict__ Y, int n_rows, float alpha, int accumulate) {
    __shared__ float sW[HID * HID];  // 16 KB

    // ---- stage W into LDS (async-to-LDS path on CDNA5 when available) ----
#if USE_ASYNC_LDS
    for (int i = threadIdx.x * 4; i < HID * HID; i += 1024) {
        __builtin_amdgcn_global_load_async_to_lds_b128(
            (__attribute__((address_space(1))) v4i_vs*)(void*)(Wg + i),
            (__attribute__((address_space(3))) v4i_vs*)(sW + i),
            /*offset=*/0, /*cpol=*/0);
    }
#if __has_builtin(__builtin_amdgcn_s_wait_asynccnt)
    __builtin_amdgcn_s_wait_asynccnt(0);
#else
    asm volatile("s_wait_asynccnt 0" ::: "memory");
#endif
#else
    for (int i = threadIdx.x * 4; i < HID * HID; i += 1024) {
        float4 v = *(const float4*)(Wg + i);
        *(float4*)(sW + i) = v;
    }
#endif
    __syncthreads();

    int wave = threadIdx.x >> 5;  // 0..7
    int lane = threadIdx.x & 31;
    int half = lane >> 4;         // 0 or 1
    int lm   = lane & 15;         // 0..15
    int row_tile = blockIdx.x * 128 + wave * 16;

    // A fragment row (clamped; garbage rows only feed unstored output rows)
    int m  = row_tile + lm;
    int mc = m < n_rows ? m : (n_rows - 1);
    const float* Xr = X + (size_t)mc * HID;

    v8f acc0 = {0.f, 0.f, 0.f, 0.f, 0.f, 0.f, 0.f, 0.f};
    v8f acc1 = acc0, acc2 = acc0, acc3 = acc0;

#pragma unroll
    for (int k = 0; k < 64; k += 4) {
        int ka = k + half * 2;  // lanes 0-15: K=k,k+1 ; lanes 16-31: K=k+2,k+3
        v2f a;
        a.x = Xr[ka];
        a.y = Xr[ka + 1];
        const float* w0 = &sW[ka * HID + lm];
        const float* w1 = &sW[(ka + 1) * HID + lm];
        v2f b0, b1, b2, b3;
        b0.x = w0[0];  b0.y = w1[0];
        b1.x = w0[16]; b1.y = w1[16];
        b2.x = w0[32]; b2.y = w1[32];
        b3.x = w0[48]; b3.y = w1[48];
        acc0 = __builtin_amdgcn_wmma_f32_16x16x4_f32(false, a, false, b0, (short)0, acc0, false, false);
        acc1 = __builtin_amdgcn_wmma_f32_16x16x4_f32(false, a, false, b1, (short)0, acc1, false, false);
        acc2 = __builtin_amdgcn_wmma_f32_16x16x4_f32(false, a, false, b2, (short)0, acc2, false, false);
        acc3 = __builtin_amdgcn_wmma_f32_16x16x4_f32(false, a, false, b3, (short)0, acc3, false, false);
    }

    // ---- epilogue: rowscale folded here (rs[m] * sum), alpha, bias/beta ----
#pragma unroll
    for (int i = 0; i < 8; ++i) {
        int row = row_tile + half * 8 + i;  // VGPR i: rows i (lo half), i+8 (hi half)
        int rc  = row < n_rows ? row : (n_rows - 1);
        float rsr = alpha * (rowscale ? rowscale[rc] : 1.0f);
        if (row < n_rows) {
            size_t base_idx = (size_t)row * HID + lm;
            float v0 = rsr * acc0[i];
            float v1 = rsr * acc1[i];
            float v2 = rsr * acc2[i];
            float v3 = rsr * acc3[i];
            float b0_ = accumulate ? Y[base_idx +  0] : (bias ? bias[lm +  0] : 0.0f);
            float b1_ = accumulate ? Y[base_idx + 16] : (bias ? bias[lm + 16] : 0.0f);
            float b2_ = accumulate ? Y[base_idx + 32] : (bias ? bias[lm + 32] : 0.0f);
            float b3_ = accumulate ? Y[base_idx + 48] : (bias ? bias[lm + 48] : 0.0f);
            Y[base_idx +  0] = b0_ + v0;
            Y[base_idx + 16] = b1_ + v1;
            Y[base_idx + 32] = b2_ + v2;
            Y[base_idx + 48] = b3_ + v3;
        }
    }
}

// ---------------------------------------------------------------------------
// GAT kernels
// ---------------------------------------------------------------------------
__global__ void gat_fold(const float* __restrict__ Wg, const float* __restrict__ a_l,
                         const float* __restrict__ a_r, float* __restrict__ wl,
                         float* __restrict__ wr) {
    int i = threadIdx.x;  // 64 threads
    float sl = 0.f, sr = 0.f;
#pragma unroll 8
    for (int j = 0; j < HID; ++j) {
        float w = Wg[i * HID + j];
        sl += w * a_l[j];
        sr += w * a_r[j];
    }
    wl[i] = sl;
    wr[i] = sr;
}

__global__ void gat_nodes(const float* __restrict__ h, const float* __restrict__ wl,
                          const float* __restrict__ wr, float* __restrict__ el,
                          float* __restrict__ er, int n) {
    int i = blockIdx.x * blockDim.x + threadIdx.x;
    if (i >= n) return;
    const float* hr = h + (size_t)i * HID;
    float sl = 0.f, sr = 0.f;
#pragma unroll 8
    for (int j = 0; j < HID; ++j) {
        float v = hr[j];
        sl += v * wl[j];
        sr += v * wr[j];
    }
    el[i] = sl;
    er[i] = sr;
}

__device__ inline void atomicMaxF(float* addr, float val) {
    unsigned int* a = (unsigned int*)addr;
    unsigned int old = *a;
    while (true) {
        float f = __uint_as_float(old);
        if (f >= val) break;
        unsigned int assumed = old;
        old = atomicCAS(a, assumed, __float_as_uint(val));
        if (old == assumed) break;
    }
}

__global__ void gat_logits(const float* __restrict__ el, const float* __restrict__ er,
                           const int* __restrict__ src, const int* __restrict__ dst,
                           float* __restrict__ ebuf, float* __restrict__ mmax, int E) {
    int e = blockIdx.x * blockDim.x + threadIdx.x;
    if (e >= E) return;
    float x = el[src[e]] + er[dst[e]];
    x = fmaxf(x, 0.2f * x);  // leaky_relu, slope 0.2
    ebuf[e] = x;
    atomicMaxF(&mmax[dst[e]], x);
}

__global__ void gat_exp(float* __restrict__ ebuf, const int* __restrict__ dst,
                        const float* __restrict__ mmax, float* __restrict__ denom, int E) {
    int e = blockIdx.x * blockDim.x + threadIdx.x;
    if (e >= E) return;
    float v = expf(ebuf[e] - mmax[dst[e]]);
    ebuf[e] = v;
    atomicAdd(&denom[dst[e]], v);
}

__global__ void gat_norm(float* __restrict__ ebuf, const int* __restrict__ dst,
                         const float* __restrict__ denom, int E) {
    int e = blockIdx.x * blockDim.x + threadIdx.x;
    if (e >= E) return;
    ebuf[e] = ebuf[e] / denom[dst[e]];
}

// ---------------------------------------------------------------------------
// Finalize: out[:U] = 0.4*ue + 0.3*h_agg + 0.3*h_soc ; out[U:U+G] = item_emb
// ---------------------------------------------------------------------------
__global__ void finalize_kernel(const float* __restrict__ ue, const float* __restrict__ ha,
                                const float* __restrict__ hs, const float* __restrict__ item,
                                float* __restrict__ out, long U, long G) {
    long total = U + G;
    for (long t = blockIdx.x * (long)blockDim.x + threadIdx.x; t < total;
         t += (long)gridDim.x * blockDim.x) {
        if (t < U)
            out[t] = 0.4f * ue[t] + 0.3f * ha[t] + 0.3f * hs[t];
        else
            out[t] = item[t - U];
    }
}

// ---------------------------------------------------------------------------
// Host orchestration
// ---------------------------------------------------------------------------
extern "C" void kernel_launch(void* const* d_in, const int* in_sizes, int n_in,
                              void* d_out, int out_size, void* d_ws, size_t ws_size,
                              hipStream_t stream) {
    const float* user_emb     = (const float*)d_in[0];
    const float* item_emb     = (const float*)d_in[1];
    const float* gu_weight    = (const float*)d_in[2];
    const float* W_item       = (const float*)d_in[3];  // [L,R,64,64]
    const float* W_self_item  = (const float*)d_in[4];
    const float* W_neigh_item = (const float*)d_in[5];
    const float* b_item       = (const float*)d_in[6];
    const float* W_gat        = (const float*)d_in[7];
    const float* a_l          = (const float*)d_in[8];
    const float* a_r          = (const float*)d_in[9];
    const float* W_self_soc   = (const float*)d_in[10];
    const float* W_neigh_soc  = (const float*)d_in[11];
    const float* b_soc        = (const float*)d_in[12];
    const int* item_src = (const int*)d_in[13];  // [R,E_ITEM]
    const int* item_dst = (const int*)d_in[14];
    const int* gu_src   = (const int*)d_in[15];
    const int* gu_dst   = (const int*)d_in[16];
    const int* soc_src  = (const int*)d_in[17];
    const int* soc_dst  = (const int*)d_in[18];

    const int NU = 100000, NG = 50000, Rr = 3, Ll = 2;
    const int EI = 500000, EG = 1000000, ES = 1000000;
    const long G = (long)NG * HID, U = (long)NU * HID;
    (void)in_sizes; (void)n_in; (void)out_size; (void)ws_size;

    // scratch layout
    float* ws = (float*)d_ws;
    size_t off = 0;
    auto alloc = [&](long n) { float* p = ws + off; off += (size_t)n; return p; };
    float* bufA     = alloc(G);
    float* bufB     = alloc(G);
    float* agg      = alloc(G);
    float* dinv     = alloc((long)Rr * 2 * NG);  // per relation: [out(src) | in(dst)]
    float* hn_gu    = alloc(U);
    float* h_agg    = alloc(U);
    float* rdeg_gu  = alloc(NU);
    float* wl       = alloc(64);
    float* wr       = alloc(64);
    float* el       = alloc(NU);
    float* er       = alloc(NU);
    float* mmax     = alloc(NU);
    float* denom    = alloc(NU);
    float* ebuf     = alloc(ES);
    float* rdeg_soc = alloc(NU);
    float* hn_soc   = alloc(U);
    float* h_soc    = alloc(U);

    const int T = 256;
    auto gb    = [&](long n) { return (unsigned)((n + T - 1) / T); };
    auto gb128 = [&](int n)  { return (unsigned)((n + 127) / 128); };

    // ---- item graph degrees: deg_out per src, deg_in per dst, -> ^(-1/2) ----
    fill_f32<<<gb((long)Rr * 2 * NG), T, 0, stream>>>(dinv, (long)Rr * 2 * NG, 0.f);
    for (int r = 0; r < Rr; ++r) {
        deg_count<<<gb(EI), T, 0, stream>>>(item_src + (long)r * EI, EI,
                                            dinv + (long)r * 2 * NG);
        deg_count<<<gb(EI), T, 0, stream>>>(item_dst + (long)r * EI, EI,
                                            dinv + (long)r * 2 * NG + NG);
    }
    recip_kernel<<<gb((long)Rr * 2 * NG), T, 0, stream>>>(dinv, Rr * 2 * NG, 1);

    // ---- stacked HeteroGraphConv on item graph ----
    const float inv_R = 1.0f / (float)Rr;
    const float* h_in = item_emb;
    float* h_out = bufA;
    for (int l = 0; l < Ll; ++l) {
        for (int r = 0; r < Rr; ++r) {
            fill_f32<<<gb(G), T, 0, stream>>>(agg, G, 0.f);
            scatter_add<<<gb((long)EI * HID), T, 0, stream>>>(
                h_in, item_src + (long)r * EI, item_dst + (long)r * EI,
                dinv + (long)r * 2 * NG, nullptr, agg, (long)EI * HID);
            gemm64_wmma<<<gb128(NG), T, 0, stream>>>(
                agg, dinv + (long)r * 2 * NG + NG,
                W_item + ((long)l * Rr + r) * HID * HID, nullptr,
                h_out, NG, inv_R, r > 0 ? 1 : 0);
        }
        h_in = h_out;
        h_out = bufB;
    }
    const float* h_game = h_in;  // == bufB after 2 layers

    // ---- edge-weighted SAGE game -> user ----
    fill_f32<<<gb(NU), T, 0, stream>>>(rdeg_gu, NU, 0.f);
    deg_count<<<gb(EG), T, 0, stream>>>(gu_dst, EG, rdeg_gu);
    recip_kernel<<<gb(NU), T, 0, stream>>>(rdeg_gu, NU, 0);
    fill_f32<<<gb(U), T, 0, stream>>>(hn_gu, U, 0.f);
    scatter_add<<<gb((long)EG * HID), T, 0, stream>>>(
        h_game, gu_src, gu_dst, nullptr, gu_weight, hn_gu, (long)EG * HID);
    gemm64_wmma<<<gb128(NU), T, 0, stream>>>(user_emb, nullptr, W_self_item, b_item,
                                             h_agg, NU, 1.0f, 0);
    gemm64_wmma<<<gb128(NU), T, 0, stream>>>(hn_gu, rdeg_gu, W_neigh_item, nullptr,
                                             h_agg, NU, 1.0f, 1);

    // ---- GAT attention over social graph (el = h @ (Wg@a_l), er = h @ (Wg@a_r)) ----
    gat_fold<<<1, 64, 0, stream>>>(W_gat, a_l, a_r, wl, wr);
    gat_nodes<<<gb(NU), T, 0, stream>>>(h_agg, wl, wr, el, er, NU);
    fill_f32<<<gb(NU), T, 0, stream>>>(mmax, NU, -3.402823466e38f);
    fill_f32<<<gb(NU), T, 0, stream>>>(denom, NU, 0.f);
    gat_logits<<<gb(ES), T, 0, stream>>>(el, er, soc_src, soc_dst, ebuf, mmax, ES);
    gat_exp<<<gb(ES), T, 0, stream>>>(ebuf, soc_dst, mmax, denom, ES);
    gat_norm<<<gb(ES), T, 0, stream>>>(ebuf, soc_dst, denom, ES);

    // ---- attention-weighted SAGE user -> user ----
    fill_f32<<<gb(NU), T, 0, stream>>>(rdeg_soc, NU, 0.f);
    deg_count<<<gb(ES), T, 0, stream>>>(soc_dst, ES, rdeg_soc);
    recip_kernel<<<gb(NU), T, 0, stream>>>(rdeg_soc, NU, 0);
    fill_f32<<<gb(U), T, 0, stream>>>(hn_soc, U, 0.f);
    scatter_add<<<gb((long)ES * HID), T, 0, stream>>>(
        user_emb, soc_src, soc_dst, nullptr, ebuf, hn_soc, (long)ES * HID);
    gemm64_wmma<<<gb128(NU), T, 0, stream>>>(user_emb, nullptr, W_self_soc, b_soc,
                                             h_soc, NU, 1.0f, 0);
    gemm64_wmma<<<gb128(NU), T, 0, stream>>>(hn_soc, rdeg_soc, W_neigh_soc, nullptr,
                                             h_soc, NU, 1.0f, 1);

    // ---- finalize ----
    finalize_kernel<<<gb(U + G), T, 0, stream>>>(user_emb, h_agg, h_soc, item_emb,
                                                 (float*)d_out, U, G);
}